// ImageBlockRotationAndTranslation_5231270167091
// MI455X (gfx1250) — compile-verified
//
#include <hip/hip_runtime.h>

typedef __attribute__((ext_vector_type(2))) float v2f;
typedef __attribute__((ext_vector_type(8))) float v8f;

#define IMG_H 192
#define IMG_W 192
#define BLK 32
#define NXB 6
#define NSH 9
#define NSHIFT 81   /* 9*9 */
#define NANG 7
#define PATCH_W 44  /* padded LDS row stride (multiple of 4 for float4 stores) */
#define PATCH_R 40  /* fixed 40x40 staged search window */
#define SROW 36     /* padded LDS row stride for 32-wide reference block */
#define NTHREADS 192

__global__ __launch_bounds__(NTHREADS)
void block_match_kernel(const float* __restrict__ fm1,
                        const float* __restrict__ fm2,
                        float* __restrict__ block_min /* [4*36] */) {
    __shared__ float4   tabW[BLK * BLK];        // bilinear corner weights (validity folded in)
    __shared__ unsigned tabC[BLK * BLK];        // packed clipped corner coords y0|x0<<8|y1<<16|x1<<24
    __shared__ float    patch[PATCH_R * PATCH_W];
    __shared__ float    sblk[BLK * SROW];
    __shared__ float    wave_min[12];

    const int tid   = threadIdx.x;
    const int lane  = tid & 31;
    const int wave  = tid >> 5;           // 0..5
    const int m     = lane & 15;          // candidate row within the 16-wide WMMA tile
    const int half  = lane >> 4;          // which pixel half of the candidate this lane covers
    const int blk   = blockIdx.x;         // 0..35
    const int batch = blockIdx.y;         // 0..3
    const int by = blk / NXB, bx = blk % NXB;

    // ---- stage i_fm1 block (float4 path, 16B aligned: col base = bx*32) ----
    for (int i = tid; i < BLK * (BLK / 4); i += NTHREADS) {
        int row = i >> 3, c4 = i & 7;
        float4 v = *(const float4*)&fm1[(batch * IMG_H + by * BLK + row) * IMG_W + bx * BLK + c4 * 4];
        *(float4*)&sblk[row * SROW + c4 * 4] = v;
    }
    // ---- stage fixed 40x40 i_fm2 search window; origin clamped so it covers all shifted patches.
    //      r0/c0 are multiples of 4 (0,28,60,92,124,152) so float4 loads stay aligned. ----
    const int r0 = min(max(by * BLK - 4, 0), IMG_H - PATCH_R);
    const int c0 = min(max(bx * BLK - 4, 0), IMG_W - PATCH_R);
    for (int i = tid; i < PATCH_R * (PATCH_R / 4); i += NTHREADS) {
        int row = i / (PATCH_R / 4), c4 = i % (PATCH_R / 4);
        float4 v = *(const float4*)&fm2[(batch * IMG_H + r0 + row) * IMG_W + c0 + c4 * 4];
        *(float4*)&patch[row * PATCH_W + c4 * 4] = v;
    }

    // This lane-pair's candidate shift (clamped for the 15 invalid slots of wave 5).
    const int shift = wave * 16 + m;
    const int sc    = min(shift, NSHIFT - 1);
    const int syi = sc / NSH, sxi = sc % NSH;
    const int ybase = min(max(by * BLK + syi - 4, 0), IMG_H - BLK) - r0;
    const int xbase = min(max(bx * BLK + sxi - 4, 0), IMG_W - BLK) - c0;

    float lmin = 3.4e38f;

    for (int a = 0; a < NANG; ++a) {
        __syncthreads();   // also covers the initial staging on a==0
        // ---- build rotation-resampling table for this angle ----
        const float rad = (float)(a - 3) * 0.017453292519943295f;
        const float ca = cosf(rad), sa = sinf(rad);
        for (int q = tid; q < BLK * BLK; q += NTHREADS) {
            int qy = q >> 5, qx = q & 31;
            float xo = -1.0f + (float)qx * (2.0f / 31.0f);
            float yo = -1.0f + (float)qy * (2.0f / 31.0f);
            float xi = ca * xo - sa * yo;
            float yi = sa * xo + ca * yo;
            float px = (xi + 1.0f) * 0.5f * 31.0f;
            float py = (yi + 1.0f) * 0.5f * 31.0f;
            float x0f = floorf(px), y0f = floorf(py);
            float fx = px - x0f, fy = py - y0f;
            int x0 = (int)x0f, y0 = (int)y0f;
            int x1 = x0 + 1,   y1 = y0 + 1;
            float vx0 = (x0 >= 0 && x0 < BLK) ? 1.0f : 0.0f;
            float vx1 = (x1 >= 0 && x1 < BLK) ? 1.0f : 0.0f;
            float vy0 = (y0 >= 0 && y0 < BLK) ? 1.0f : 0.0f;
            float vy1 = (y1 >= 0 && y1 < BLK) ? 1.0f : 0.0f;
            float w00 = (1.0f - fy) * (1.0f - fx) * vy0 * vx0;
            float w01 = (1.0f - fy) * fx          * vy0 * vx1;
            float w10 = fy          * (1.0f - fx) * vy1 * vx0;
            float w11 = fy          * fx          * vy1 * vx1;
            int x0c = min(max(x0, 0), BLK - 1), y0c = min(max(y0, 0), BLK - 1);
            int x1c = min(max(x1, 0), BLK - 1), y1c = min(max(y1, 0), BLK - 1);
            tabW[q] = make_float4(w00, w01, w10, w11);
            tabC[q] = (unsigned)y0c | ((unsigned)x0c << 8) | ((unsigned)y1c << 16) | ((unsigned)x1c << 24);
        }
        __syncthreads();

        // ---- accumulate weighted SSD (num) and mask sum (den) over this lane's pixel quarters ----
        v2f accN = {0.0f, 0.0f};
        v2f accD = {0.0f, 0.0f};
        for (int k = 0; k < 2; ++k) {
            float an = 0.0f, ad = 0.0f;
            const int qstart = half * 512 + k * 256;
            for (int i = 0; i < 256; ++i) {
                const int q = qstart + i;
                const float4   wv  = tabW[q];
                const unsigned cpk = tabC[q];
                const int y0c =  cpk        & 255;
                const int x0c = (cpk >> 8)  & 255;
                const int y1c = (cpk >> 16) & 255;
                const int x1c = (cpk >> 24) & 255;
                const float* prow0 = &patch[(ybase + y0c) * PATCH_W + xbase];
                const float* prow1 = &patch[(ybase + y1c) * PATCH_W + xbase];
                float r = wv.x * prow0[x0c] + wv.y * prow0[x1c]
                        + wv.z * prow1[x0c] + wv.w * prow1[x1c];
                float wsum = (wv.x + wv.y) + (wv.z + wv.w);
                float d = sblk[(q >> 5) * SROW + (q & 31)] - r;
                an = fmaf(wsum * d, d, an);
                ad += wsum;
            }
            accN[k] = an;
            accD[k] = ad;
        }

        // ---- WMMA cross-lane reduction: A(16x4 partials) x B(ones in column N=0) ----
        // A layout (32b A 16x4): lanes0-15 hold K=0,1; lanes16-31 hold K=2,3 -> matches accN/accD.
        v2f bones;
        bones.x = (m == 0) ? 1.0f : 0.0f;   // B[k][0] = 1 for all k, else 0
        bones.y = bones.x;
        v8f zero = {};
        v8f num = __builtin_amdgcn_wmma_f32_16x16x4_f32(false, accN, false, bones,
                                                        (short)0, zero, false, false);
        v8f den = __builtin_amdgcn_wmma_f32_16x16x4_f32(false, accD, false, bones,
                                                        (short)0, zero, false, false);
        // den is the angle's mask sum: identical for every candidate row (shift-independent),
        // so one reciprocal covers all 8 rows. Computed on all lanes (1/0 = inf on lanes
        // whose N-column is zero; harmless) to keep the division out of divergent control flow.
        float inv = 1.0f / den[0];
        // D[m][0] lives on lane0 (rows 0-7) and lane16 (rows 8-15), VGPR j = row j (+8).
        if (m == 0) {
            #pragma unroll
            for (int j = 0; j < 8; ++j) {
                int sh = wave * 16 + half * 8 + j;
                float mse = (sh < NSHIFT) ? num[j] * inv : 3.4e38f;  // branch-free select
                lmin = fminf(lmin, mse);
            }
        }
    }

    if (m == 0) wave_min[wave * 2 + half] = lmin;
    __syncthreads();
    if (tid == 0) {
        float bm = wave_min[0];
        #pragma unroll
        for (int i = 1; i < 12; ++i) bm = fminf(bm, wave_min[i]);
        block_min[batch * 36 + blk] = bm;
    }
}

__global__ void sum_min_kernel(const float* __restrict__ block_min, float* __restrict__ out) {
    int b = threadIdx.x;
    if (b < 4) {
        float s = 0.0f;
        for (int i = 0; i < 36; ++i) s += block_min[b * 36 + i];   // fixed order: deterministic
        out[b] = s;
    }
}

extern "C" void kernel_launch(void* const* d_in, const int* in_sizes, int n_in,
                              void* d_out, int out_size, void* d_ws, size_t ws_size,
                              hipStream_t stream) {
    const float* fm1 = (const float*)d_in[0];
    const float* fm2 = (const float*)d_in[1];
    float* out = (float*)d_out;
    float* ws  = (float*)d_ws;   // 144 floats of per-block minima

    dim3 grid(36, 4);
    block_match_kernel<<<grid, NTHREADS, 0, stream>>>(fm1, fm2, ws);
    sum_min_kernel<<<1, 32, 0, stream>>>(ws, out);
}